// MahalanobisLayer_5437428597311
// MI455X (gfx1250) — compile-verified
//
#include <hip/hip_runtime.h>

// ---------------------------------------------------------------------------
// Mahalanobis row-wise quadratic form: out[i] = (x-x_fit)[i] @ S_inv @ (x-x_fit)[i]^T
// N=65536, D=256, fp32. Compute-bound in fp32 matrix ops -> V_WMMA_F32_16X16X4_F32.
// ---------------------------------------------------------------------------

typedef float v2f __attribute__((ext_vector_type(2)));
typedef float v8f __attribute__((ext_vector_type(8)));

#define NROWS 65536
#define DDIM  256
#define ROW_STRIDE 260                 // padded LDS row stride (floats): bank-conflict-free + 16B aligned
#define PANEL_FLOATS (16 * ROW_STRIDE) // one 16-row delta panel per wave
#define WAVES_PER_BLOCK 8

__launch_bounds__(256)
__global__ void mahalanobis_wmma_kernel(const float* __restrict__ x,
                                        const float* __restrict__ xfit,
                                        const float* __restrict__ Sinv,
                                        float* __restrict__ out) {
    extern __shared__ float lds[];

    const int lane  = threadIdx.x & 31;
    const int wave  = threadIdx.x >> 5;
    const int n     = lane & 15;   // N-column within tile / row selector for A
    const int khalf = lane >> 4;   // 0: K even pair base, 1: K odd pair base (per ISA A/B layout)

    const int tile  = blockIdx.x * WAVES_PER_BLOCK + wave;  // 16-row tile index
    const size_t base = (size_t)tile * 16;                  // first global row

    float* panel = lds + wave * PANEL_FLOATS;

    // ---- Stage delta = x - x_fit (16 x 256) into this wave's LDS panel -----
    // 4096 floats = 1024 float4; 32 lanes x 32 iters, fully coalesced.
    const float* xp  = x    + base * DDIM;
    const float* xfp = xfit + base * DDIM;
#pragma unroll
    for (int t = 0; t < 32; ++t) {
        const int i = (t * 32 + lane) * 4;     // flat float index in 16x256 tile
        const int r = i >> 8;                  // row 0..15
        const int c = i & 255;                 // col 0..255 (multiple of 4)
        float4 a = *(const float4*)(xp  + i);
        float4 b = *(const float4*)(xfp + i);
        float4 d = make_float4(a.x - b.x, a.y - b.y, a.z - b.z, a.w - b.w);
        *(float4*)(panel + r * ROW_STRIDE + c) = d;
    }
    // Fence LDS stores before fragment reads (same-wave; CDNA5 split counter).
    asm volatile("s_wait_dscnt 0" ::: "memory");

    // ---- T = Delta @ S_inv, accumulated 16x16 tiles, K swept 4 at a time ----
    // B fragment: need S_inv[k][jcol]; S_inv is symmetric, so read S_inv[jcol][k]
    // which is contiguous in k -> 8-byte loads per lane (hot in WGP$/L2).
    float outAcc[8];
#pragma unroll
    for (int r = 0; r < 8; ++r) outAcc[r] = 0.0f;

#pragma unroll
    for (int jg = 0; jg < 4; ++jg) {           // groups of 4 column tiles: reuse A frag 4x
        v8f acc[4];
#pragma unroll
        for (int jj = 0; jj < 4; ++jj) acc[jj] = (v8f)(0.0f);

        const float* brow0 = Sinv + (size_t)((jg * 4 + 0) * 16 + n) * DDIM;
        const float* brow1 = Sinv + (size_t)((jg * 4 + 1) * 16 + n) * DDIM;
        const float* brow2 = Sinv + (size_t)((jg * 4 + 2) * 16 + n) * DDIM;
        const float* brow3 = Sinv + (size_t)((jg * 4 + 3) * 16 + n) * DDIM;

#pragma unroll 4
        for (int kk = 0; kk < 64; ++kk) {
            const int k = kk * 4 + khalf * 2;  // A/B frag: VGPR0 holds K=2*half, VGPR1 K=2*half+1
            v2f a  = *(const v2f*)(panel + n * ROW_STRIDE + k);
            v2f b0 = *(const v2f*)(brow0 + k);
            v2f b1 = *(const v2f*)(brow1 + k);
            v2f b2 = *(const v2f*)(brow2 + k);
            v2f b3 = *(const v2f*)(brow3 + k);
            acc[0] = __builtin_amdgcn_wmma_f32_16x16x4_f32(false, a, false, b0, (short)0, acc[0], false, false);
            acc[1] = __builtin_amdgcn_wmma_f32_16x16x4_f32(false, a, false, b1, (short)0, acc[1], false, false);
            acc[2] = __builtin_amdgcn_wmma_f32_16x16x4_f32(false, a, false, b2, (short)0, acc[2], false, false);
            acc[3] = __builtin_amdgcn_wmma_f32_16x16x4_f32(false, a, false, b3, (short)0, acc[3], false, false);
        }

        // Fold T-tile elementwise with delta: outAcc[r] += T[M][n] * delta[M][j*16+n]
        // C/D layout: VGPR r <-> M = 8*khalf + r, N = lane&15.
#pragma unroll
        for (int jj = 0; jj < 4; ++jj) {
            const int j = jg * 4 + jj;
#pragma unroll
            for (int r = 0; r < 8; ++r) {
                const int row = khalf * 8 + r;
                outAcc[r] += acc[jj][r] * panel[row * ROW_STRIDE + j * 16 + n];
            }
        }
    }

    // ---- Reduce across the 16 lanes holding the same matrix row ------------
    // offsets 1,2,4,8 stay within each 16-lane half (lanes 0-15: M=r, 16-31: M=8+r)
#pragma unroll
    for (int r = 0; r < 8; ++r) {
        float v = outAcc[r];
        v += __shfl_xor(v, 1, 32);
        v += __shfl_xor(v, 2, 32);
        v += __shfl_xor(v, 4, 32);
        v += __shfl_xor(v, 8, 32);
        outAcc[r] = v;
    }

    if (n == 0) {  // lane 0 -> rows base..base+7, lane 16 -> rows base+8..base+15
#pragma unroll
        for (int r = 0; r < 8; ++r) {
            out[base + khalf * 8 + r] = outAcc[r];
        }
    }
}

extern "C" void kernel_launch(void* const* d_in, const int* in_sizes, int n_in,
                              void* d_out, int out_size, void* d_ws, size_t ws_size,
                              hipStream_t stream) {
    const float* x    = (const float*)d_in[0];  // [N, D]
    const float* xfit = (const float*)d_in[1];  // [N, D]
    const float* Sinv = (const float*)d_in[2];  // [D, D], symmetric
    float* out = (float*)d_out;                 // [N]

    const int rows_per_block = WAVES_PER_BLOCK * 16;          // 128
    dim3 grid(NROWS / rows_per_block);                        // 512 blocks
    dim3 block(256);                                          // 8 wave32
    size_t ldsBytes = (size_t)WAVES_PER_BLOCK * PANEL_FLOATS * sizeof(float); // 133120 B

    mahalanobis_wmma_kernel<<<grid, block, ldsBytes, stream>>>(x, xfit, Sinv, out);
}